// SDHLoss_52398601011946
// MI455X (gfx1250) — compile-verified
//
#include <hip/hip_runtime.h>

#define N_ROWS 4096
#define NBIT   64
#define NCLS   100
#define LMBD0  0.001f
#define LMBD1  1.0f
#define LMBD2  0.001f
#define ALPHA  1.0f
#define EPSI   1e-7f

typedef float v2f __attribute__((ext_vector_type(2)));
typedef float v8f __attribute__((ext_vector_type(8)));

// workspace layout (floats):
//  [0] = sum_i q1_i   (J1 numerator)
//  [1] = R = sum_i r_i
//  [2] = raw j3 sum = sum (ortho-1)^2
//  [WS_CNT  .. +100) = cnt[c]
//  [WS_RSUM .. +100) = rsum[c]
//  [WS_S    .. +6400) = s[c][e]  class sums of tanh(u)
#define WS_CNT   16
#define WS_RSUM  128
#define WS_S     256
#define WS_TOTAL (WS_S + NCLS * NBIT)

__global__ void sdh_zero_ws(float* __restrict__ ws) {
    int i = blockIdx.x * 256 + threadIdx.x;
    if (i < WS_TOTAL) ws[i] = 0.0f;
}

__device__ __forceinline__ float sgnf(float x) {
    return (x > 0.0f) ? 1.0f : ((x < 0.0f) ? -1.0f : 0.0f);
}

// One row (64 floats) per 16 lanes; each lane owns a float4. 2 rows per wave.
__global__ void sdh_row_stats(const float* __restrict__ u,
                              const float* __restrict__ y,
                              float* __restrict__ ws) {
    const int tid    = threadIdx.x;
    const int lane   = tid & 31;
    const int half   = lane >> 4;     // which row inside the wave
    const int part   = lane & 15;     // float4 slot inside the row
    const int waveId = (blockIdx.x * blockDim.x + tid) >> 5;
    const int row    = waveId * 2 + half;
    if (row >= N_ROWS) return;

    const float4 uv = *(const float4*)(u + row * NBIT + part * 4);
    float4 t;
    t.x = tanhf(uv.x); t.y = tanhf(uv.y); t.z = tanhf(uv.z); t.w = tanhf(uv.w);

    float dx = t.x - sgnf(t.x), dy = t.y - sgnf(t.y);
    float dz = t.z - sgnf(t.z), dw = t.w - sgnf(t.w);
    float q1 = dx*dx + dy*dy + dz*dz + dw*dw;
    float r  = t.x*t.x + t.y*t.y + t.z*t.z + t.w*t.w;

    // half-wave (16-lane) reductions; xor offsets 1..8 stay within the half
    #pragma unroll
    for (int off = 1; off < 16; off <<= 1) {
        q1 += __shfl_xor(q1, off, 32);
        r  += __shfl_xor(r,  off, 32);
    }

    // label = index of the one-hot column; lanes scan strided columns
    int lbl = -1;
    for (int c = part; c < NCLS; c += 16)
        if (y[row * NCLS + c] > 0.5f) lbl = c;
    #pragma unroll
    for (int off = 1; off < 16; off <<= 1) {
        int o = __shfl_xor(lbl, off, 32);
        lbl = (o > lbl) ? o : lbl;
    }

    // per-class feature sums: each lane adds its float4
    float* sc = ws + WS_S + lbl * NBIT + part * 4;
    atomicAdd(sc + 0, t.x);
    atomicAdd(sc + 1, t.y);
    atomicAdd(sc + 2, t.z);
    atomicAdd(sc + 3, t.w);

    if (part == 0) {
        atomicAdd(ws + WS_CNT  + lbl, 1.0f);
        atomicAdd(ws + WS_RSUM + lbl, r);
        atomicAdd(ws + 0, q1);
        atomicAdd(ws + 1, r);
    }
}

// j3 = sum((W W^T - 1)^2) via V_WMMA_F32_16X16X4_F32. One wave per 16x16 tile,
// 7x7 tiles cover 100x100. Out-of-range tile rows are CLAMPED (not guarded):
// their data can only reach D entries with m>=100 or n>=100, which are masked
// out of the reduction below — so the K-loop is fully branchless.
__global__ void __launch_bounds__(32)
sdh_j3_wmma(const float* __restrict__ w, float* __restrict__ ws) {
    const int lane  = threadIdx.x & 31;
    const int ti    = blockIdx.x / 7;
    const int tj    = blockIdx.x % 7;
    const int part  = lane & 15;
    const int khalf = (lane >> 4) * 2;   // A/B lane layout: K pair selector

    int arow = ti * 16 + part;           // A: m = lane&15
    int brow = tj * 16 + part;           // B (=W^T cols): n = lane&15
    const int ac = (arow < NCLS) ? arow : (NCLS - 1);   // clamp, always valid
    const int bc = (brow < NCLS) ? brow : (NCLS - 1);
    const float* ap = w + ac * NBIT + khalf;
    const float* bp = w + bc * NBIT + khalf;

    // Hoist all operand loads: 16 b64 loads each for A and B, no EXEC games.
    v2f A[16], B[16];
    #pragma unroll
    for (int i = 0; i < 16; ++i) {
        A[i] = *(const v2f*)(ap + 4 * i);
        B[i] = *(const v2f*)(bp + 4 * i);
    }

    // K=64 as 16 chained fp32 WMMAs, back-to-back on the matrix pipe.
    v8f acc = {};
    #pragma unroll
    for (int i = 0; i < 16; ++i) {
        acc = __builtin_amdgcn_wmma_f32_16x16x4_f32(
            false, A[i], false, B[i], (short)0, acc, false, false);
    }

    // C/D layout: reg v, lane l -> m = v + 8*(l>>4), n = l&15
    float psum = 0.0f;
    const int n = tj * 16 + part;
    #pragma unroll
    for (int v = 0; v < 8; ++v) {
        int m = ti * 16 + v + 8 * (lane >> 4);
        if (m < NCLS && n < NCLS) {
            float d = acc[v] - 1.0f;
            psum += d * d;
        }
    }
    #pragma unroll
    for (int off = 16; off > 0; off >>= 1)
        psum += __shfl_xor(psum, off, 32);
    if (lane == 0) atomicAdd(ws + 2, psum);
}

__global__ void sdh_finalize(const float* __restrict__ ws, float* __restrict__ out) {
    __shared__ float red[256];
    const int tid = threadIdx.x;
    const float* s = ws + WS_S;

    // SS = sum_c ||s_c||^2  (sum of squares of all s elements)
    float v = 0.0f;
    for (int i = tid; i < NCLS * NBIT; i += 256) { float x = s[i]; v += x * x; }
    red[tid] = v; __syncthreads();
    for (int st = 128; st > 0; st >>= 1) { if (tid < st) red[tid] += red[tid + st]; __syncthreads(); }
    float SS = red[0]; __syncthreads();

    // T2 = || sum_c s_c ||^2
    v = 0.0f;
    if (tid < NBIT) {
        float te = 0.0f;
        for (int c = 0; c < NCLS; ++c) te += s[c * NBIT + tid];
        v = te * te;
    }
    red[tid] = v; __syncthreads();
    for (int st = 128; st > 0; st >>= 1) { if (tid < st) red[tid] += red[tid + st]; __syncthreads(); }
    float T2 = red[0]; __syncthreads();

    // POSN = sum cnt^2
    v = (tid < NCLS) ? ws[WS_CNT + tid] * ws[WS_CNT + tid] : 0.0f;
    red[tid] = v; __syncthreads();
    for (int st = 128; st > 0; st >>= 1) { if (tid < st) red[tid] += red[tid + st]; __syncthreads(); }
    float POSN = red[0]; __syncthreads();

    // CR = sum cnt[c]*rsum[c]
    v = (tid < NCLS) ? ws[WS_CNT + tid] * ws[WS_RSUM + tid] : 0.0f;
    red[tid] = v; __syncthreads();
    for (int st = 128; st > 0; st >>= 1) { if (tid < st) red[tid] += red[tid + st]; __syncthreads(); }
    float CR = red[0];

    if (tid == 0) {
        const float Nf   = (float)N_ROWS;
        float j1   = ws[0] / Nf;                 // mean row quantization loss
        float R    = ws[1];                      // j2_1 = trace(u^T u)
        float Spos = 2.0f * CR - 2.0f * SS;      // sum(d2 * sim)
        float Sall = 2.0f * Nf * R - 2.0f * T2;  // sum(d2)
        float Sneg = Sall - Spos;
        float negn = Nf * Nf - POSN;
        float trp  = Spos / (POSN + EPSI);
        float trn  = Sneg / (negn + EPSI);
        float j2   = (R + ALPHA * (trn - trp)) * 0.5f;
        float j3   = ws[2] * 0.5f;
        out[0] = LMBD0 * j1 - LMBD1 * j2 + LMBD2 * j3;
    }
}

extern "C" void kernel_launch(void* const* d_in, const int* in_sizes, int n_in,
                              void* d_out, int out_size, void* d_ws, size_t ws_size,
                              hipStream_t stream) {
    const float* u  = (const float*)d_in[0];   // (4096, 64)
    const float* y  = (const float*)d_in[1];   // (4096, 100) one-hot
    const float* w  = (const float*)d_in[2];   // (100, 64)
    float* out = (float*)d_out;
    float* ws  = (float*)d_ws;

    (void)in_sizes; (void)n_in; (void)out_size; (void)ws_size;

    sdh_zero_ws<<<(WS_TOTAL + 255) / 256, 256, 0, stream>>>(ws);

    // 4096 rows, 2 rows/wave, 8 waves/block -> 256 blocks of 256 threads
    sdh_row_stats<<<N_ROWS / 16, 256, 0, stream>>>(u, y, ws);

    // 7x7 tiles of 16x16 over the 100x100 Gram matrix, one wave per tile
    sdh_j3_wmma<<<49, 32, 0, stream>>>(w, ws);

    sdh_finalize<<<1, 256, 0, stream>>>(ws, out);
}